// GCN2_58789512348197
// MI455X (gfx1250) — compile-verified
//
#include <hip/hip_runtime.h>
#include <math.h>

// CDNA5 / gfx1250. fp32 WMMA (v_wmma_f32_16x16x4_f32) for all dense GEMMs,
// native f32 global atomics for the SpMM scatter (L2-resident on 192MB L2).

typedef __attribute__((ext_vector_type(2))) float v2f;
typedef __attribute__((ext_vector_type(8))) float v8f;

#define NFEAT 512
#define NHID  128
#define NCLS  40

// ---------------- zero fill (graph-capture-safe memset) ----------------
__global__ void GCN_fill0(float* __restrict__ p, long n) {
  long i = (long)blockIdx.x * blockDim.x + threadIdx.x;
  if (i < n) p[i] = 0.0f;
}

// ---------------- GEMM1: out[N,128] = x[N,512] @ W[512,128] ----------------
// One wave per 16-row tile; 8 accumulators cover all 128 output columns.
// A (16x4 f32): lane<16 -> K=0,1 ; lane>=16 -> K=2,3 (M = lane&15)
// B (4x16 f32): mirrored, N = lane&15
// C/D (16x16 f32): element r of lane -> M = r + 8*(lane>>4), N = lane&15
__global__ __launch_bounds__(256) void GCN_gemm_xw(
    const float* __restrict__ x, const float* __restrict__ W,
    float* __restrict__ out, int nTilesM) {
  const int lane = threadIdx.x & 31;
  const int wid  = threadIdx.x >> 5;
  const int mtile = blockIdx.x * 8 + wid;
  if (mtile >= nTilesM) return;              // wave-granular exit: EXEC stays all-1s
  const int row = lane & 15;
  const int hi  = lane >> 4;
  const int m0  = mtile * 16;
  const float* __restrict__ xrow = x + (long)(m0 + row) * NFEAT;

  v8f acc[8] = {};
  for (int k = 0; k < NFEAT; k += 4) {
    const int ka = k + 2 * hi;
    v2f a;
    a.x = xrow[ka];
    a.y = xrow[ka + 1];
    const float* __restrict__ w0 = W + (long)ka * NHID;       // row ka
    const float* __restrict__ w1 = w0 + NHID;                 // row ka+1
#pragma unroll
    for (int t = 0; t < 8; ++t) {
      const int col = t * 16 + row;
      v2f b;
      b.x = w0[col];
      b.y = w1[col];
      acc[t] = __builtin_amdgcn_wmma_f32_16x16x4_f32(
          false, a, false, b, (short)0, acc[t], false, false);
    }
  }
#pragma unroll
  for (int t = 0; t < 8; ++t)
#pragma unroll
    for (int r = 0; r < 8; ++r)
      out[(long)(m0 + r + 8 * hi) * NHID + t * 16 + row] = acc[t][r];
}

// ------- GEMM2: out[N,40] = relu(agg[N,128] + bias) @ W[128,40] (padded to 48) -------
__global__ __launch_bounds__(256) void GCN_gemm_h(
    const float* __restrict__ agg, const float* __restrict__ bias,
    const float* __restrict__ W, float* __restrict__ out, int nTilesM) {
  const int lane = threadIdx.x & 31;
  const int wid  = threadIdx.x >> 5;
  const int mtile = blockIdx.x * 8 + wid;
  if (mtile >= nTilesM) return;
  const int row = lane & 15;
  const int hi  = lane >> 4;
  const int m0  = mtile * 16;
  const float* __restrict__ arow = agg + (long)(m0 + row) * NHID;

  v8f acc[3] = {};
  for (int k = 0; k < NHID; k += 4) {
    const int ka = k + 2 * hi;
    v2f a;
    a.x = fmaxf(arow[ka] + bias[ka], 0.0f);          // fused bias + relu
    a.y = fmaxf(arow[ka + 1] + bias[ka + 1], 0.0f);
#pragma unroll
    for (int t = 0; t < 3; ++t) {
      const int col = t * 16 + row;
      v2f b;
      b.x = (col < NCLS) ? W[(long)ka * NCLS + col] : 0.0f;
      b.y = (col < NCLS) ? W[(long)(ka + 1) * NCLS + col] : 0.0f;
      acc[t] = __builtin_amdgcn_wmma_f32_16x16x4_f32(
          false, a, false, b, (short)0, acc[t], false, false);
    }
  }
#pragma unroll
  for (int t = 0; t < 3; ++t) {
    const int col = t * 16 + row;
    if (col < NCLS)
#pragma unroll
      for (int r = 0; r < 8; ++r)
        out[(long)(m0 + r + 8 * hi) * NCLS + col] = acc[t][r];
  }
}

// ---------------- SpMM (128 feats): one wave per edge, float4 per lane ----------------
__global__ void GCN_spmm128(const int* __restrict__ src, const int* __restrict__ dst,
                            const float* __restrict__ vals,
                            const float* __restrict__ h, float* __restrict__ out, int E) {
  const int e = (int)(((long)blockIdx.x * blockDim.x + threadIdx.x) >> 5);
  const int lane = threadIdx.x & 31;
  if (e >= E) return;
  const int s = src[e], d = dst[e];
  const float v = vals[e];
  const float4 hv = *(const float4*)(h + (long)s * NHID + lane * 4);
  float* o = out + (long)d * NHID + lane * 4;
  unsafeAtomicAdd(o + 0, v * hv.x);
  unsafeAtomicAdd(o + 1, v * hv.y);
  unsafeAtomicAdd(o + 2, v * hv.z);
  unsafeAtomicAdd(o + 3, v * hv.w);
}

// ------- SpMM (40 feats): one thread per (edge, float2 pair), 20 pairs/edge -------
// Full lane utilization; the 20 threads of an edge are adjacent so src/dst/val
// loads coalesce to a single request and the float2 gathers are contiguous.
__global__ void GCN_spmm40(const int* __restrict__ src, const int* __restrict__ dst,
                           const float* __restrict__ vals,
                           const float* __restrict__ h, float* __restrict__ out,
                           long npairs) {
  const long t = (long)blockIdx.x * blockDim.x + threadIdx.x;
  if (t >= npairs) return;
  const int e = (int)(t / 20);        // constant division -> mul/shift
  const int p = (int)(t - (long)e * 20);
  const int s = src[e], d = dst[e];
  const float v = vals[e];
  const float2 hv = *(const float2*)(h + (long)s * NCLS + p * 2);
  float* o = out + (long)d * NCLS + p * 2;
  unsafeAtomicAdd(o + 0, v * hv.x);
  unsafeAtomicAdd(o + 1, v * hv.y);
}

// ---------------- gated fusion + log_softmax, one thread per node ----------------
__global__ void GCN_fuse(const float* __restrict__ aggo1, const float* __restrict__ aggo2,
                         const float* __restrict__ b2, const float* __restrict__ b4,
                         const float* __restrict__ Wl, const float* __restrict__ bl,
                         float* __restrict__ out, int Nn) {
  const int n = blockIdx.x * blockDim.x + threadIdx.x;
  if (n >= Nn) return;
  float o1[NCLS], o2[NCLS], z[NCLS];
  for (int j = 0; j < NCLS; ++j) {
    o1[j] = aggo1[(long)n * NCLS + j] + b2[j];
    o2[j] = aggo2[(long)n * NCLS + j] + b4[j];
  }
  for (int j = 0; j < NCLS; ++j) {
    float g = bl[j];
    for (int i = 0; i < NCLS; ++i) g += o1[i] * Wl[i * NCLS + j];
    for (int i = 0; i < NCLS; ++i) g += o2[i] * Wl[(NCLS + i) * NCLS + j];
    g = 1.0f / (1.0f + expf(-g));            // sigmoid
    z[j] = g * o1[j] + (1.0f - g) * o2[j];
  }
  float mx = z[0];
  for (int j = 1; j < NCLS; ++j) mx = fmaxf(mx, z[j]);
  float s = 0.0f;
  for (int j = 0; j < NCLS; ++j) s += expf(z[j] - mx);
  const float lse = mx + logf(s);
  for (int j = 0; j < NCLS; ++j) out[(long)n * NCLS + j] = z[j] - lse;
}

extern "C" void kernel_launch(void* const* d_in, const int* in_sizes, int n_in,
                              void* d_out, int out_size, void* d_ws, size_t ws_size,
                              hipStream_t stream) {
  const float* x   = (const float*)d_in[0];
  const int*   ei1 = (const int*)  d_in[1];
  const float* ev1 = (const float*)d_in[2];
  const int*   ei2 = (const int*)  d_in[3];
  const float* ev2 = (const float*)d_in[4];
  const float* W1 = (const float*)d_in[5];  const float* b1 = (const float*)d_in[6];
  const float* W2 = (const float*)d_in[7];  const float* b2 = (const float*)d_in[8];
  const float* W3 = (const float*)d_in[9];  const float* b3 = (const float*)d_in[10];
  const float* W4 = (const float*)d_in[11]; const float* b4 = (const float*)d_in[12];
  const float* Wl = (const float*)d_in[13]; const float* bl = (const float*)d_in[14];
  float* out = (float*)d_out;

  const int E  = in_sizes[2];           // 800000
  const int Nn = in_sizes[0] / NFEAT;   // 50000
  const int nTilesM = (Nn + 15) / 16;   // 3125 (exact)

  // Workspace layout (floats). Total = 4 * Nn * NHID = 25.6M floats = 102.4 MB.
  float* ws = (float*)d_ws;
  const size_t SZH = (size_t)Nn * NHID;     // 6.4M floats
  const size_t SZC = (size_t)Nn * NCLS;     // 2.0M floats
  float* AGG1 = ws;                         // [Nn,128]
  float* AGG2 = ws + SZH;                   // [Nn,128]
  float* XW1  = ws + 2 * SZH;               // [Nn,128]
  float* XW3  = ws + 3 * SZH;               // [Nn,128]
  // After spmm1 consumes XW1/XW3, reuse that region:
  float* HW1   = XW1;                       // [Nn,40]
  float* HW2   = XW1 + SZC;                 // [Nn,40]
  float* AGGO1 = XW1 + 2 * SZC;             // [Nn,40]
  float* AGGO2 = XW1 + 3 * SZC;             // [Nn,40]

  const dim3 gblk((nTilesM + 7) / 8);
  const int spmm128Blocks = (int)(((long)E * 32 + 255) / 256);
  const long npairs = (long)E * 20;
  const int spmm40Blocks = (int)((npairs + 255) / 256);

  // 1. zero layer-1 accumulators (AGG1|AGG2 contiguous)
  { long n = 2L * SZH; GCN_fill0<<<(n + 255) / 256, 256, 0, stream>>>(AGG1, n); }
  // 2. dense GEMM xW for both towers (x stays L2-resident across the two passes)
  GCN_gemm_xw<<<gblk, 256, 0, stream>>>(x, W1, XW1, nTilesM);
  GCN_gemm_xw<<<gblk, 256, 0, stream>>>(x, W3, XW3, nTilesM);
  // 3. layer-1 sparse aggregate
  GCN_spmm128<<<spmm128Blocks, 256, 0, stream>>>(ei1, ei1 + E, ev1, XW1, AGG1, E);
  GCN_spmm128<<<spmm128Blocks, 256, 0, stream>>>(ei2, ei2 + E, ev2, XW3, AGG2, E);
  // 4. relu(agg+b) @ W2/W4 (fused bias+relu into the WMMA A-operand load)
  GCN_gemm_h<<<gblk, 256, 0, stream>>>(AGG1, b1, W2, HW1, nTilesM);
  GCN_gemm_h<<<gblk, 256, 0, stream>>>(AGG2, b3, W4, HW2, nTilesM);
  // 5. zero layer-2 accumulators (AGGO1|AGGO2 contiguous)
  { long n = 2L * SZC; GCN_fill0<<<(n + 255) / 256, 256, 0, stream>>>(AGGO1, n); }
  // 6. layer-2 sparse aggregate
  GCN_spmm40<<<spmm40Blocks, 256, 0, stream>>>(ei1, ei1 + E, ev1, HW1, AGGO1, npairs);
  GCN_spmm40<<<spmm40Blocks, 256, 0, stream>>>(ei2, ei2 + E, ev2, HW2, AGGO2, npairs);
  // 7. gated fusion + log_softmax
  GCN_fuse<<<(Nn + 255) / 256, 256, 0, stream>>>(AGGO1, AGGO2, b2, b4, Wl, bl, out, Nn);
}